// TealActor_36464272343129
// MI455X (gfx1250) — compile-verified
//
#include <hip/hip_runtime.h>

#define NN   1000000
#define PN   500000
#define NNZ_ 16000000

typedef __attribute__((ext_vector_type(2))) float v2f;
typedef __attribute__((ext_vector_type(4))) float v4f;
typedef __attribute__((ext_vector_type(8))) float v8f;
typedef __attribute__((ext_vector_type(4))) int   v4i;

// ---------------------------------------------------------------------------
// Per-node tiny linear: out[n,:] = W * [prev_row(n), h0[n]] + b   (W is [D,D])
// prev rows come from bufS (non-path) or bufD (path), width D-1.
// ---------------------------------------------------------------------------
template<int D>
__global__ void gnn_linear(const float* __restrict__ prevS,
                           const float* __restrict__ prevD,
                           const float* __restrict__ feat,
                           const float* __restrict__ W,
                           const float* __restrict__ bias,
                           float* __restrict__ out) {
    int n = blockIdx.x * blockDim.x + threadIdx.x;
    if (n >= NN) return;
    float x[D];
    if constexpr (D > 1) {
        const float* src = (n < NN - PN) ? (prevS + (size_t)n * (D - 1))
                                         : (prevD + (size_t)(n - (NN - PN)) * (D - 1));
        #pragma unroll
        for (int t = 0; t < D - 1; ++t) x[t] = src[t];
    }
    x[D - 1] = feat[n];
    float* orow = out + (size_t)n * D;
    #pragma unroll
    for (int k = 0; k < D; ++k) {
        float acc = bias[k];
        #pragma unroll
        for (int t = 0; t < D; ++t) acc += W[k * D + t] * x[t];
        orow[k] = acc;
    }
}

// ---------------------------------------------------------------------------
// SpMM scatter: hs[row[e],:] += val[e] * hl[col[e],:]
// 4 edges/thread, b128 non-temporal edge loads (stream, don't pollute L2 --
// node buffers stay L2-resident for the random gathers), hardware f32 atomic
// add (no-return, device scope) forced via inline asm.
// ---------------------------------------------------------------------------
template<int D>
__global__ void spmm_scatter(const int* __restrict__ row,
                             const int* __restrict__ col,
                             const float* __restrict__ val,
                             const float* __restrict__ hl,
                             float* __restrict__ hs) {
    int e4 = blockIdx.x * blockDim.x + threadIdx.x;
    if (e4 >= NNZ_ / 4) return;
    v4i r = __builtin_nontemporal_load((const v4i*)row + e4);
    v4i c = __builtin_nontemporal_load((const v4i*)col + e4);
    v4f v = __builtin_nontemporal_load((const v4f*)val + e4);
    #pragma unroll
    for (int j = 0; j < 4; ++j) {
        const float* src = hl + (size_t)c[j] * D;
        float* dst = hs + (size_t)r[j] * D;
        #pragma unroll
        for (int k = 0; k < D; ++k) {
            float prod = v[j] * src[k];
            asm volatile("global_atomic_add_f32 %0, %1, off scope:SCOPE_DEV"
                         :: "v"(dst + k), "v"(prod) : "memory");
        }
    }
}

// ---------------------------------------------------------------------------
// Path-node DNN via fp32 WMMA 16x16x4:  O = X * W^T + b
// X = [100000, K=5D] (flat view of path rows), W = [K,K], O = [100000, K].
// One wave per 16x16 output tile; K looped in steps of 4.  Out-of-range K
// elements (last step only) are handled branchlessly: clamp the index so the
// load is always legal, then select 0 -- compiles to v_cndmask, no EXEC
// manipulation, so EXEC is all-1s through every WMMA.
// A layout (32-bit 16x4): lane m=lane&15; VGPR0 -> K = kb + 2*(lane>>4),
//                         VGPR1 -> K = kb + 1 + 2*(lane>>4).
// B layout (32-bit 4x16): lane n=lane&15; same K striping across VGPRs.
// C/D layout: c[r] = O[16*mt + r + 8*(lane>>4), n].
// ---------------------------------------------------------------------------
template<int D>
__global__ __launch_bounds__(64) void dnn_wmma(const float* __restrict__ S,
                                               const float* __restrict__ W,
                                               const float* __restrict__ Bv,
                                               float* __restrict__ O) {
    constexpr int K  = 5 * D;
    constexpr int JT = (K + 15) / 16;
    const int lane = threadIdx.x & 31;
    const int wave = blockIdx.x * 2 + (threadIdx.x >> 5);
    const int jt = wave % JT;
    const int mt = wave / JT;          // 0..6249 (exact grid sizing)
    const int li = lane & 15;
    const int hi = lane >> 4;
    const int n = jt * 16 + li;        // output / B column
    const int m = mt * 16 + li;        // A row
    const bool ncol = (n < K);
    const int  nc   = ncol ? n : (K - 1);   // clamped (always-legal) column

    float bias = Bv[nc];
    bias = ncol ? bias : 0.0f;
    v8f c;
    #pragma unroll
    for (int r = 0; r < 8; ++r) c[r] = bias;

    const float* Srow = S + (size_t)m * K;
    const float* Wrow = W + (size_t)nc * K;
    #pragma unroll
    for (int kb = 0; kb < K; kb += 4) {
        const int k0  = kb + 2 * hi;
        const int k1  = k0 + 1;
        const int k0c = (k0 < K) ? k0 : 0;
        const int k1c = (k1 < K) ? k1 : 0;
        float a0 = Srow[k0c], a1 = Srow[k1c];
        float b0 = Wrow[k0c], b1 = Wrow[k1c];
        v2f a, b;
        a.x = (k0 < K) ? a0 : 0.0f;
        a.y = (k1 < K) ? a1 : 0.0f;
        b.x = (ncol && k0 < K) ? b0 : 0.0f;
        b.y = (ncol && k1 < K) ? b1 : 0.0f;
        c = __builtin_amdgcn_wmma_f32_16x16x4_f32(false, a, false, b,
                                                  (short)0, c, false, false);
    }
    if (ncol) {
        #pragma unroll
        for (int r = 0; r < 8; ++r) {
            int mr = mt * 16 + r + 8 * hi;
            O[(size_t)mr * K + n] = c[r];
        }
    }
}

// ---------------------------------------------------------------------------
// Final readout: mean[q,:] = mean_w * x[q,:] + mean_b, x[q,7j+t] from bufD/feat
// ---------------------------------------------------------------------------
__global__ void mean_kernel(const float* __restrict__ hD,   // [PN, 6]
                            const float* __restrict__ feat,
                            const float* __restrict__ mw,   // [5, 35]
                            const float* __restrict__ mb,   // [5]
                            float* __restrict__ out) {
    int q = blockIdx.x * blockDim.x + threadIdx.x;
    if (q >= PN / 5) return;
    float x[35];
    #pragma unroll
    for (int j = 0; j < 5; ++j) {
        const float* dr = hD + (size_t)(5 * q + j) * 6;
        #pragma unroll
        for (int t = 0; t < 6; ++t) x[7 * j + t] = dr[t];
        x[7 * j + 6] = feat[(NN - PN) + 5 * q + j];
    }
    #pragma unroll
    for (int o = 0; o < 5; ++o) {
        float acc = mb[o];
        #pragma unroll
        for (int k = 0; k < 35; ++k) acc += mw[o * 35 + k] * x[k];
        out[(size_t)q * 5 + o] = acc;
    }
}

__global__ void fill_std(float* out, int start, int total) {
    int i = start + blockIdx.x * blockDim.x + threadIdx.x;
    if (i < total) out[i] = 1.0f;
}

// ---------------------------------------------------------------------------
extern "C" void kernel_launch(void* const* d_in, const int* in_sizes, int n_in,
                              void* d_out, int out_size, void* d_ws, size_t ws_size,
                              hipStream_t stream) {
    (void)in_sizes; (void)n_in; (void)ws_size;
    const float* feature = (const float*)d_in[0];
    const int*   erow    = (const int*)d_in[1];          // edge_index[0]
    const int*   ecol    = erow + NNZ_;                  // edge_index[1]
    const float* evals   = (const float*)d_in[2];
    const float* gnn_w[6]; const float* gnn_b[6];
    const float* dnn_w[6]; const float* dnn_b[6];
    for (int i = 0; i < 6; ++i) {
        gnn_w[i] = (const float*)d_in[3 + i];
        gnn_b[i] = (const float*)d_in[9 + i];
        dnn_w[i] = (const float*)d_in[15 + i];
        dnn_b[i] = (const float*)d_in[21 + i];
    }
    const float* mean_w = (const float*)d_in[27];
    const float* mean_b = (const float*)d_in[28];

    float* bufL = (float*)d_ws;                    // [NN, 6]  linear output
    float* bufS = bufL + (size_t)NN * 6;           // [NN, 6]  spmm output
    float* bufD = bufS + (size_t)NN * 6;           // [PN, 6]  dnn output (path)
    float* outp = (float*)d_out;

#define RUN_LAYER(i, D)                                                                  \
    do {                                                                                 \
        gnn_linear<D><<<(NN + 255) / 256, 256, 0, stream>>>(                             \
            bufS, bufD, feature, gnn_w[i], gnn_b[i], bufL);                              \
        hipMemsetAsync(bufS, 0, (size_t)NN * (D) * sizeof(float), stream);               \
        spmm_scatter<D><<<(NNZ_ / 4 + 255) / 256, 256, 0, stream>>>(                     \
            erow, ecol, evals, bufL, bufS);                                              \
        constexpr int jt##i = ((5 * (D)) + 15) / 16;                                     \
        dnn_wmma<D><<<(6250 * jt##i) / 2, 64, 0, stream>>>(                              \
            bufS + (size_t)(NN - PN) * (D), dnn_w[i], dnn_b[i], bufD);                   \
    } while (0)

    RUN_LAYER(0, 1);
    RUN_LAYER(1, 2);
    RUN_LAYER(2, 3);
    RUN_LAYER(3, 4);
    RUN_LAYER(4, 5);
    RUN_LAYER(5, 6);
#undef RUN_LAYER

    mean_kernel<<<(PN / 5 + 255) / 256, 256, 0, stream>>>(bufD, feature, mean_w, mean_b, outp);
    if (out_size > PN) {
        int rem = out_size - PN;
        fill_std<<<(rem + 255) / 256, 256, 0, stream>>>(outp, PN, out_size);
    }
}